// TreeLSTMforNLI_29480655519923
// MI455X (gfx1250) — compile-verified
//
#include <hip/hip_runtime.h>
#include <math.h>

typedef float v2f __attribute__((ext_vector_type(2)));
typedef float v8f __attribute__((ext_vector_type(8)));

constexpr int H    = 300;      // hidden / input dim
constexpr int NCLS = 3;
constexpr int B    = 64;
constexpr int T    = 128;
constexpr int B2   = 2 * B;        // both sentences: 128
constexpr int HP   = 320;          // padded hidden (20 x 16 tiles)
constexpr int M_ALL= B2 * T;       // 16384 rows
constexpr int NG   = 4 * HP;       // 1280 gate columns (i|f|o|u)
constexpr int KWH  = 4 * H;        // 1200 (classifier K)

__device__ __forceinline__ float sigf(float x) { return 1.0f / (1.0f + expf(-x)); }

// ---------------------------------------------------------------------------
// Pack / pad all weight matrices into WMMA-friendly padded layouts.
// W4p  : (NG x HP)  row-major [n][k], gate g rows at n = g*HP (+bias fold in b4p)
// b4p  : (NG)       gate bias; i/o/u gates get u*_b folded in (hj == 0 insight)
// Ufp  : (HP x HP)  uf_w padded, row-major [n][k]
// ufbp : (HP)
// Gwp  : (HP x HP)  gcn_w padded, row-major [k][n]  (B used K-major)
// Whp  : (HP x KWH) wh_w padded rows, row-major [n][k]
// whbp : (HP)
// ---------------------------------------------------------------------------
__global__ void pack_params(
    const float* wi_w, const float* wi_b, const float* wf_w, const float* wf_b,
    const float* wo_w, const float* wo_b, const float* wu_w, const float* wu_b,
    const float* ui_b, const float* uf_w, const float* uf_b,
    const float* uo_b, const float* uu_b,
    const float* gcn_w, const float* wh_w, const float* wh_b,
    float* W4p, float* b4p, float* Ufp, float* ufbp,
    float* Gwp, float* Whp, float* whbp)
{
  const long nW4 = (long)NG * HP;
  const long nUf = (long)HP * HP;
  const long nGw = nUf;
  const long nWh = (long)HP * KWH;
  const long total = nW4 + NG + nUf + HP + nGw + nWh + HP;
  long i = (long)blockIdx.x * blockDim.x + threadIdx.x;
  const long stride = (long)gridDim.x * blockDim.x;
  for (; i < total; i += stride) {
    long r = i;
    if (r < nW4) {
      int n = (int)(r / HP), k = (int)(r % HP);
      int gg = n / HP;  int rr = n % HP;
      float v = 0.f;
      if (rr < H && k < H) {
        const float* w = (gg == 0) ? wi_w : (gg == 1) ? wf_w : (gg == 2) ? wo_w : wu_w;
        v = w[rr * H + k];
      }
      W4p[r] = v;
      continue;
    }
    r -= nW4;
    if (r < NG) {
      int n = (int)r; int gg = n / HP, rr = n % HP;
      float v = 0.f;
      if (rr < H) {
        if      (gg == 0) v = wi_b[rr] + ui_b[rr];
        else if (gg == 1) v = wf_b[rr];
        else if (gg == 2) v = wo_b[rr] + uo_b[rr];
        else              v = wu_b[rr] + uu_b[rr];
      }
      b4p[n] = v;
      continue;
    }
    r -= NG;
    if (r < nUf) {
      int n = (int)(r / HP), k = (int)(r % HP);
      Ufp[r] = (n < H && k < H) ? uf_w[n * H + k] : 0.f;
      continue;
    }
    r -= nUf;
    if (r < HP) { ufbp[r] = (r < H) ? uf_b[r] : 0.f; continue; }
    r -= HP;
    if (r < nGw) {
      int k = (int)(r / HP), n = (int)(r % HP);
      Gwp[r] = (k < H && n < H) ? gcn_w[k * H + n] : 0.f;
      continue;
    }
    r -= nGw;
    if (r < nWh) {
      int n = (int)(r / KWH), k = (int)(r % KWH);
      Whp[r] = (n < H) ? wh_w[n * KWH + k] : 0.f;
      continue;
    }
    r -= nWh;
    whbp[r] = (r < H) ? wh_b[r] : 0.f;
  }
}

// Embedding gather into padded X matrix: rows m = b2*T + t (b2<64 left, else right)
__global__ void gather_embed(const int* __restrict__ linp, const int* __restrict__ rinp,
                             const float* __restrict__ emb, float* __restrict__ Xg)
{
  const long total = (long)M_ALL * HP;
  long i = (long)blockIdx.x * blockDim.x + threadIdx.x;
  const long stride = (long)gridDim.x * blockDim.x;
  for (; i < total; i += stride) {
    int m = (int)(i / HP), col = (int)(i % HP);
    int b2 = m >> 7, t = m & (T - 1);
    int tok = (b2 < B) ? linp[b2 * T + t] : rinp[(b2 - B) * T + t];
    Xg[i] = (col < H) ? emb[(long)tok * H + col] : 0.f;
  }
}

// ---------------------------------------------------------------------------
// Generic f32 WMMA GEMM:  C[m,n] = sum_k A[m,k] * B(k,n)  (+ bias[n]) (+relu)
// BT=1: Bm stored (N x K) row-major (B(k,n)=Bm[n*ldb+k]); BT=0: (K x N).
// Block = 256 threads = 8 waves arranged 2(M) x 4(N); block tile 32 x 64.
// Uses V_WMMA_F32_16X16X4_F32 (exact f32 math, matches reference precision).
// ---------------------------------------------------------------------------
template<int BT, int EPI>
__global__ __launch_bounds__(256) void gemm_wmma(
    const float* __restrict__ A, int lda,
    const float* __restrict__ Bm, int ldb,
    float* __restrict__ C, int ldc,
    const float* __restrict__ bias, int K)
{
  const int lane = threadIdx.x & 31;
  const int wave = threadIdx.x >> 5;
  const int tileM = blockIdx.x * 32 + (wave >> 2) * 16;
  const int tileN = blockIdx.y * 64 + (wave & 3) * 16;
  const int r  = lane & 15;
  const int hi = lane >> 4;
  v8f acc = {};
  const float* arow = A + (size_t)(tileM + r) * lda;
  for (int kk = 0; kk < K; kk += 4) {
    const int ko = kk + hi * 2;
    v2f a = *(const v2f*)(arow + ko);
    v2f b;
    if (BT) {
      b = *(const v2f*)(Bm + (size_t)(tileN + r) * ldb + ko);
    } else {
      b.x = Bm[(size_t)ko * ldb + tileN + r];
      b.y = Bm[(size_t)(ko + 1) * ldb + tileN + r];
    }
    acc = __builtin_amdgcn_wmma_f32_16x16x4_f32(false, a, false, b,
                                                (short)0, acc, false, false);
  }
  for (int e = 0; e < 8; ++e) {
    const int mrow = tileM + hi * 8 + e;
    const int ncol = tileN + r;
    float v = acc[e] + (bias ? bias[ncol] : 0.f);
    if (EPI == 1) v = fmaxf(v, 0.f);
    C[(size_t)mrow * ldc + ncol] = v;
  }
}

// Fuse i/o/u gates in place (hj == 0 => no recurrence):
//   slot0 <- sigmoid(Xi)*tanh(Xu)   (i*u),  slot2 <- sigmoid(Xo),  slot1 = Xf kept
__global__ void fuse_gates(float* __restrict__ X4)
{
  const long total = (long)M_ALL * HP;
  long i = (long)blockIdx.x * blockDim.x + threadIdx.x;
  const long stride = (long)gridDim.x * blockDim.x;
  for (; i < total; i += stride) {
    int m = (int)(i / HP), col = (int)(i % HP);
    float* row = X4 + (size_t)m * NG;
    float xi = row[col];
    float xo = row[2 * HP + col];
    float xu = row[3 * HP + col];
    row[col]          = sigf(xi) * tanhf(xu);
    row[2 * HP + col] = sigf(xo);
  }
}

// ---------------------------------------------------------------------------
// Sequential tree-LSTM scan. 8 workgroups x 16 batch rows; 128 steps.
// Phase 1 (VALU): c[j] = i*u + sum_{graph(j,t)!=0} sigmoid(Xf[j]+F[t]) * c[t]
//                 h[j] = o * tanh(c[j])      (h_new staged to LDS)
// Phase 2 (WMMA): F[j] = h[j] @ uf_w^T + uf_b   (M=16, N=320, K=320, f32 WMMA)
// ---------------------------------------------------------------------------
__global__ __launch_bounds__(256) void scan_lstm(
    const float* __restrict__ lgraph, const float* __restrict__ rgraph,
    const float* __restrict__ X4, const float* __restrict__ Ufp,
    const float* __restrict__ ufbp,
    float* __restrict__ hbuf, float* __restrict__ cbuf, float* __restrict__ Fbuf)
{
  __shared__ float sh_h[16][HP];
  __shared__ float sh_g[16][T];
  const int grp  = blockIdx.x;              // 0..7 -> b2 in [16g, 16g+16)
  const int tid  = threadIdx.x;
  const int lane = tid & 31;
  const int wave = tid >> 5;
  const int r  = lane & 15;
  const int hi = lane >> 4;

  for (int j = 0; j < T; ++j) {
    // stage graph row j for the 16 batch rows
    for (int idx = tid; idx < 16 * T; idx += 256) {
      int bl = idx >> 7, t = idx & (T - 1);
      int b2 = grp * 16 + bl;
      const float* gp = (b2 < B) ? (lgraph + (size_t)b2 * T * T)
                                 : (rgraph + (size_t)(b2 - B) * T * T);
      sh_g[bl][t] = gp[j * T + t];
    }
    __syncthreads();

    // phase 1: each wave owns 2 batch rows -> graph branches wave-uniform
    for (int bl = wave * 2; bl < wave * 2 + 2; ++bl) {
      const int b2 = grp * 16 + bl;
      const size_t mj = (size_t)b2 * T + j;
      const float* xrow = X4 + mj * NG;
      for (int col = lane; col < HP; col += 32) {
        float acc = xrow[col];                // i*u
        const float xf = xrow[HP + col];      // Xf (wf_b included)
        for (int t = 0; t < j; ++t) {
          const float gv = sh_g[bl][t];
          if (gv != 0.f) {
            const size_t mt = (size_t)b2 * T + t;
            acc += gv * sigf(xf + Fbuf[mt * HP + col]) * cbuf[mt * HP + col];
          }
        }
        const float o  = xrow[2 * HP + col];
        const float cn = acc;
        const float hn = (col < H) ? o * tanhf(cn) : 0.f;
        cbuf[mj * HP + col] = cn;
        hbuf[mj * HP + col] = hn;
        sh_h[bl][col] = hn;
      }
    }
    __threadfence_block();
    __syncthreads();

    // phase 2: F[j] = sh_h @ Ufp^T + ufbp via f32 WMMA (20 N-tiles over 8 waves)
    for (int nt = wave; nt < HP / 16; nt += 8) {
      const int n0 = nt * 16;
      v8f acc = {};
      for (int kk = 0; kk < HP; kk += 4) {
        const int ko = kk + hi * 2;
        v2f a; a.x = sh_h[r][ko]; a.y = sh_h[r][ko + 1];
        v2f b = *(const v2f*)(Ufp + (size_t)(n0 + r) * HP + ko);
        acc = __builtin_amdgcn_wmma_f32_16x16x4_f32(false, a, false, b,
                                                    (short)0, acc, false, false);
      }
      for (int e = 0; e < 8; ++e) {
        const int bl = hi * 8 + e;
        const int b2 = grp * 16 + bl;
        const int n  = n0 + r;
        Fbuf[((size_t)b2 * T + j) * HP + n] = acc[e] + ufbp[n];
      }
    }
    __threadfence_block();
    __syncthreads();
  }
}

// GCN aggregate: G[b,s,:] = relu(S[b,s,:] + sum_t (g[s,t]+g[t,s]) * S[b,t,:])
__global__ __launch_bounds__(320) void gcn_agg(
    const float* __restrict__ lgraph, const float* __restrict__ rgraph,
    const float* __restrict__ S, float* __restrict__ G)
{
  __shared__ float wrow[T];
  const int m  = blockIdx.x;               // 0..16383
  const int b2 = m >> 7, s = m & (T - 1);
  const float* gp = (b2 < B) ? (lgraph + (size_t)b2 * T * T)
                             : (rgraph + (size_t)(b2 - B) * T * T);
  if (threadIdx.x < T) {
    const int t = threadIdx.x;
    wrow[t] = gp[s * T + t] + gp[t * T + s];
  }
  __syncthreads();
  const int col = threadIdx.x;
  float acc = S[(size_t)m * HP + col];
  for (int t = 0; t < T; ++t) {
    const float w = wrow[t];
    if (w != 0.f) acc += w * S[((size_t)b2 * T + t) * HP + col];
  }
  G[(size_t)m * HP + col] = fmaxf(acc, 0.f);
}

// v = [ls.max | rs.max | ls.mean | rs.mean]  (B x 1200, dense for classifier GEMM)
__global__ __launch_bounds__(320) void pool_kernel(const float* __restrict__ G,
                                                   float* __restrict__ v)
{
  const int b = blockIdx.x, col = threadIdx.x;
  if (col >= H) return;
  float mxl = -1e30f, sml = 0.f, mxr = -1e30f, smr = 0.f;
  for (int s = 0; s < T; ++s) {
    const float gl = G[((size_t)b * T + s) * HP + col];
    const float gr = G[((size_t)(b + B) * T + s) * HP + col];
    mxl = fmaxf(mxl, gl); sml += gl;
    mxr = fmaxf(mxr, gr); smr += gr;
  }
  float* vb = v + (size_t)b * KWH;
  vb[col]         = mxl;
  vb[H + col]     = mxr;
  vb[2 * H + col] = sml * (1.f / T);
  vb[3 * H + col] = smr * (1.f / T);
}

// Final projection: out[b,c] = hid[b,:300] . wp_w[c,:] + wp_b[c]
__global__ void wp_out(const float* __restrict__ hid, const float* __restrict__ wp_w,
                       const float* __restrict__ wp_b, float* __restrict__ out)
{
  const int i = blockIdx.x * blockDim.x + threadIdx.x;
  if (i >= B * NCLS) return;
  const int b = i / NCLS, cls = i % NCLS;
  float acc = wp_b[cls];
  for (int k = 0; k < H; ++k) acc += hid[(size_t)b * HP + k] * wp_w[cls * H + k];
  out[i] = acc;
}

extern "C" void kernel_launch(void* const* d_in, const int* in_sizes, int n_in,
                              void* d_out, int out_size, void* d_ws, size_t ws_size,
                              hipStream_t stream)
{
  const int*   linputs = (const int*)  d_in[0];
  const float* lgraph  = (const float*)d_in[1];
  const int*   rinputs = (const int*)  d_in[2];
  const float* rgraph  = (const float*)d_in[3];
  const float* emb     = (const float*)d_in[4];
  const float* wi_w = (const float*)d_in[5],  *wi_b = (const float*)d_in[6];
  const float* wf_w = (const float*)d_in[7],  *wf_b = (const float*)d_in[8];
  const float* wo_w = (const float*)d_in[9],  *wo_b = (const float*)d_in[10];
  const float* wu_w = (const float*)d_in[11], *wu_b = (const float*)d_in[12];
  const float* ui_b = (const float*)d_in[14];
  const float* uf_w = (const float*)d_in[15], *uf_b = (const float*)d_in[16];
  const float* uo_b = (const float*)d_in[18];
  const float* uu_b = (const float*)d_in[20];
  const float* gcn_w = (const float*)d_in[21];
  const float* wh_w  = (const float*)d_in[22], *wh_b = (const float*)d_in[23];
  const float* wp_w  = (const float*)d_in[24], *wp_b = (const float*)d_in[25];

  float* ws = (float*)d_ws;
  size_t o = 0;
  float* Xg   = ws + o; o += (size_t)M_ALL * HP;   // reused as h after X GEMM
  float* hb   = Xg;
  float* W4p  = ws + o; o += (size_t)NG * HP;
  float* b4p  = ws + o; o += NG;
  float* Ufp  = ws + o; o += (size_t)HP * HP;
  float* ufbp = ws + o; o += HP;
  float* Gwp  = ws + o; o += (size_t)HP * HP;
  float* Whp  = ws + o; o += (size_t)HP * KWH;
  float* whbp = ws + o; o += HP;
  float* X4   = ws + o; o += (size_t)M_ALL * NG;   // reused: S at base, G after
  float* Sb   = X4;
  float* Gb   = X4 + (size_t)M_ALL * HP;
  float* cb   = ws + o; o += (size_t)M_ALL * HP;
  float* Fb   = ws + o; o += (size_t)M_ALL * HP;
  float* vb   = ws + o; o += (size_t)B * KWH;
  float* hid  = ws + o; o += (size_t)B * HP;

  pack_params<<<1024, 256, 0, stream>>>(
      wi_w, wi_b, wf_w, wf_b, wo_w, wo_b, wu_w, wu_b,
      ui_b, uf_w, uf_b, uo_b, uu_b, gcn_w, wh_w, wh_b,
      W4p, b4p, Ufp, ufbp, Gwp, Whp, whbp);

  gather_embed<<<4096, 256, 0, stream>>>(linputs, rinputs, emb, Xg);

  // X4 = Xg @ W4p^T + b4p   (M=16384, N=1280, K=320)
  gemm_wmma<1, 0><<<dim3(M_ALL / 32, NG / 64), 256, 0, stream>>>(
      Xg, HP, W4p, HP, X4, NG, b4p, HP);

  fuse_gates<<<4096, 256, 0, stream>>>(X4);

  scan_lstm<<<8, 256, 0, stream>>>(lgraph, rgraph, X4, Ufp, ufbp, hb, cb, Fb);

  // S = h @ gcn_w   (M=16384, N=320, K=320)
  gemm_wmma<0, 0><<<dim3(M_ALL / 32, HP / 64), 256, 0, stream>>>(
      hb, HP, Gwp, HP, Sb, HP, nullptr, HP);

  gcn_agg<<<M_ALL, 320, 0, stream>>>(lgraph, rgraph, Sb, Gb);

  pool_kernel<<<B, 320, 0, stream>>>(Gb, vb);

  // hid = relu(v @ wh_w^T + wh_b)   (M=64, N=320, K=1200)
  gemm_wmma<1, 1><<<dim3(B / 32, HP / 64), 256, 0, stream>>>(
      vb, KWH, Whp, KWH, hid, HP, whbp, KWH);

  wp_out<<<1, 256, 0, stream>>>(hid, wp_w, wp_b, (float*)d_out);
}